// PruneLoraLayer_11055245819948
// MI455X (gfx1250) — compile-verified
//
#include <hip/hip_runtime.h>

typedef float v2f __attribute__((ext_vector_type(2)));
typedef float v8f __attribute__((ext_vector_type(8)));
typedef int   v4i __attribute__((ext_vector_type(4)));

typedef __attribute__((address_space(1))) v4i* global_v4i_p;
typedef __attribute__((address_space(3))) v4i* shared_v4i_p;

#define M_TOT   8192      // B*S = 4*2048
#define K_DIM   4096      // d_in
#define N_DIM   4096      // d_out
#define R_DIM   16        // lora rank
#define LORA_SCALE 2.0f

#define BM 128
#define BN 128
#define BK 32
#define LDST 36           // padded LDS row stride (floats): conflict-free frag reads, 16B-aligned rows
#define LDT  18           // padded stride for the t (=x@A^T) staging buffer

#if defined(__AMDGCN__) && __has_builtin(__builtin_amdgcn_global_load_async_to_lds_b128)
#define HAVE_ASYNC_LDS 1
#else
#define HAVE_ASYNC_LDS 0
#endif

__device__ __forceinline__ void copy16_to_lds(const float* g, float* l) {
#if HAVE_ASYNC_LDS
    // GLOBAL_LOAD_ASYNC_TO_LDS_B128: memory -> LDS without VGPR round-trip (ASYNCcnt)
    __builtin_amdgcn_global_load_async_to_lds_b128(
        (global_v4i_p)(v4i*)g,      // AS1 int4*
        (shared_v4i_p)(v4i*)l,      // AS3 int4* (addrspacecast)
        /*offset=*/0, /*cpol=*/0);
#else
    *(float4*)l = *(const float4*)g;
#endif
}

__device__ __forceinline__ void wait_async_copies() {
#if HAVE_ASYNC_LDS
#if __has_builtin(__builtin_amdgcn_s_wait_asynccnt)
    __builtin_amdgcn_s_wait_asynccnt(0);
#else
    asm volatile("s_wait_asynccnt 0x0" ::: "memory");
#endif
#endif
}

__global__ __launch_bounds__(256) void lora_linear_wmma_f32(
    const float* __restrict__ x,     // [M_TOT, K_DIM]
    const float* __restrict__ W,     // [N_DIM, K_DIM]
    const float* __restrict__ lA,    // [R_DIM, K_DIM]
    const float* __restrict__ lB,    // [N_DIM, R_DIM]
    const float* __restrict__ bias,  // [N_DIM]
    float* __restrict__ out)         // [M_TOT, N_DIM]
{
    __shared__ float smx[2][BM * LDST];
    __shared__ float smw[2][BN * LDST];

    const int tid  = threadIdx.x;
    const int lane = tid & 31;
    const int wid  = tid >> 5;
    const int hb   = lane >> 4;   // half-wave select
    const int ln   = lane & 15;

    const int wm = wid >> 2;      // 0..1 -> 64 rows per wave
    const int wn = wid & 3;       // 0..3 -> 32 cols per wave

    const int mBase = blockIdx.y * BM;
    const int nBase = blockIdx.x * BN;

    v8f acc[4][2];                // 4 row-frags x 2 col-frags of 16x16 f32
    v8f tacc[4];                  // t = x @ lora_A^T accumulators (16 cols = rank)
#pragma unroll
    for (int rm = 0; rm < 4; ++rm) {
        tacc[rm] = (v8f){0.f,0.f,0.f,0.f,0.f,0.f,0.f,0.f};
#pragma unroll
        for (int cn = 0; cn < 2; ++cn)
            acc[rm][cn] = (v8f){0.f,0.f,0.f,0.f,0.f,0.f,0.f,0.f};
    }

    const int ldRow = tid >> 3;         // 0..31
    const int ldCol = (tid & 7) << 2;   // 0,4,...,28

    // Stage one K-chunk of x and W into LDS buffer `buf` (async, b128 granules).
    auto stage = [&](int buf, int k0) {
#pragma unroll
        for (int it = 0; it < 4; ++it) {
            const int r = ldRow + it * 32;
            copy16_to_lds(&x[(size_t)(mBase + r) * K_DIM + k0 + ldCol],
                          &smx[buf][r * LDST + ldCol]);
            copy16_to_lds(&W[(size_t)(nBase + r) * K_DIM + k0 + ldCol],
                          &smw[buf][r * LDST + ldCol]);
        }
    };

    stage(0, 0);                  // prime the pipeline
    int buf = 0;

    for (int k0 = 0; k0 < K_DIM; k0 += BK) {
        wait_async_copies();      // my chunk-k0 copies have landed in LDS
        __syncthreads();          // everyone's have; prev compute done -> other buffer is free

        if (k0 + BK < K_DIM)      // issue next chunk while we compute on this one
            stage(buf ^ 1, k0 + BK);

        const float* bx = smx[buf];
        const float* bw = smw[buf];

#pragma unroll
        for (int kk = 0; kk < BK; kk += 4) {
            const int kf = kk + 2 * hb;   // A/B fragment k-offset for this lane

            v2f a[4], b[2], bA;
#pragma unroll
            for (int rm = 0; rm < 4; ++rm)
                a[rm] = *(const v2f*)&bx[(wm * 64 + rm * 16 + ln) * LDST + kf];
#pragma unroll
            for (int cn = 0; cn < 2; ++cn)
                b[cn] = *(const v2f*)&bw[(wn * 32 + cn * 16 + ln) * LDST + kf];
            // lora_A is tiny (256 KB) and hot in L2: read B-fragment directly from global
            bA = *(const v2f*)&lA[(size_t)ln * K_DIM + k0 + kf];

#pragma unroll
            for (int rm = 0; rm < 4; ++rm) {
#pragma unroll
                for (int cn = 0; cn < 2; ++cn)
                    acc[rm][cn] = __builtin_amdgcn_wmma_f32_16x16x4_f32(
                        false, a[rm], false, b[cn], (short)0, acc[rm][cn], false, false);
                tacc[rm] = __builtin_amdgcn_wmma_f32_16x16x4_f32(
                    false, a[rm], false, bA, (short)0, tacc[rm], false, false);
            }
        }
        buf ^= 1;
    }

    // ---- LoRA rank-16 correction: acc += (2 * t) @ lora_B^T ----
    __syncthreads();              // done reading LDS tiles; safe to repurpose
    float* smt = smx[0];          // 128 rows x 16 cols, stride LDT
    if (wn == 0) {
#pragma unroll
        for (int rm = 0; rm < 4; ++rm)
#pragma unroll
            for (int v = 0; v < 8; ++v)
                smt[(wm * 64 + rm * 16 + v + 8 * hb) * LDT + ln] = LORA_SCALE * tacc[rm][v];
    }
    __syncthreads();

#pragma unroll
    for (int kk = 0; kk < R_DIM; kk += 4) {
        const int kf = kk + 2 * hb;
        v2f bB[2];
#pragma unroll
        for (int cn = 0; cn < 2; ++cn)
            bB[cn] = *(const v2f*)&lB[(size_t)(nBase + wn * 32 + cn * 16 + ln) * R_DIM + kf];
#pragma unroll
        for (int rm = 0; rm < 4; ++rm) {
            v2f a = *(const v2f*)&smt[(wm * 64 + rm * 16 + ln) * LDT + kf];
#pragma unroll
            for (int cn = 0; cn < 2; ++cn)
                acc[rm][cn] = __builtin_amdgcn_wmma_f32_16x16x4_f32(
                    false, a, false, bB[cn], (short)0, acc[rm][cn], false, false);
        }
    }

    // ---- bias + store (half-wave writes 16 consecutive columns: coalesced 64B) ----
    float bv[2];
#pragma unroll
    for (int cn = 0; cn < 2; ++cn)
        bv[cn] = bias[nBase + wn * 32 + cn * 16 + ln];

#pragma unroll
    for (int rm = 0; rm < 4; ++rm) {
        const int rowBase = mBase + wm * 64 + rm * 16 + 8 * hb;
#pragma unroll
        for (int cn = 0; cn < 2; ++cn) {
            const int col = nBase + wn * 32 + cn * 16 + ln;
#pragma unroll
            for (int v = 0; v < 8; ++v)
                out[(size_t)(rowBase + v) * N_DIM + col] = acc[rm][cn][v] + bv[cn];
        }
    }
}

extern "C" void kernel_launch(void* const* d_in, const int* in_sizes, int n_in,
                              void* d_out, int out_size, void* d_ws, size_t ws_size,
                              hipStream_t stream) {
    (void)in_sizes; (void)n_in; (void)out_size; (void)d_ws; (void)ws_size;
    const float* x    = (const float*)d_in[0];
    const float* W    = (const float*)d_in[1];
    const float* lA   = (const float*)d_in[2];
    const float* lB   = (const float*)d_in[3];
    const float* bias = (const float*)d_in[4];
    float* out = (float*)d_out;

    dim3 grid(N_DIM / BN, M_TOT / BM);   // (32, 64)
    dim3 block(256);
    lora_linear_wmma_f32<<<grid, block, 0, stream>>>(x, W, lA, lB, bias, out);
}